// DWT_20976620274173
// MI455X (gfx1250) — compile-verified
//
#include <hip/hip_runtime.h>

// ---------------------------------------------------------------------------
// Haar DWT, x:(8,32,512,512) f32 -> (ll,lh,hl,hh) each (8,32,256,256) f32.
// Pure streaming op: 512 MB traffic, ~0.5 FLOP/byte -> HBM-bound (~22us at
// 23.3 TB/s). CDNA5 strategy: double-buffered GLOBAL_LOAD_ASYNC_TO_LDS_B128
// pipeline (ASYNCcnt partial waits overlap DMA of tile k+1 with the butterfly
// of tile k), conflict-free ds_load_b128 gathers, non-temporal coalesced
// stores (single-touch data must not thrash the 192MB L2).
// ---------------------------------------------------------------------------

typedef float v4f __attribute__((ext_vector_type(4)));
typedef float v2f __attribute__((ext_vector_type(2)));

#define AS_GLOBAL __attribute__((address_space(1)))
#define AS_LDS    __attribute__((address_space(3)))

// Exact pointee type of the async-LDS builtin params (per hipcc diagnostic:
// "__attribute__((__vector_size__(4 * sizeof(int)))) int __device__ *").
typedef int v4i_mem __attribute__((vector_size(16)));
typedef AS_GLOBAL v4i_mem* g_b128_ptr;
typedef AS_LDS    v4i_mem* l_b128_ptr;

#if defined(__has_builtin)
#if __has_builtin(__builtin_amdgcn_global_load_async_to_lds_b128)
#define HAVE_ASYNC_LDS 1
#endif
#endif

// Elements per output subband: 8*32*256*256
#define SUB_ELEMS  16777216u
// Elements per (b,c) input plane: 512*512; planes = 8*32 = 256

#ifdef HAVE_ASYNC_LDS

template <int N>
__device__ __forceinline__ void wait_asynccnt() {
#if __has_builtin(__builtin_amdgcn_s_wait_asynccnt)
  __builtin_amdgcn_s_wait_asynccnt(N);
#else
  asm volatile("s_wait_asynccnt %0" :: "n"(N) : "memory");
#endif
}

// ---------------------------------------------------------------------------
// Each block: 4 consecutive 16-row x 512-col input tiles of one (b,c) plane,
// double-buffered through 2 x 32 KB LDS buffers. Per tile per thread:
//   - 8 async B128 global->LDS loads (no VGPR return slots, ASYNCcnt-tracked)
//   - 8 ds_load_b128 (lane-contiguous 16B stride -> conflict-free)
//   - 16 b64 NT stores (4 subbands x 4 rows, fully coalesced)
// ---------------------------------------------------------------------------
__global__ __launch_bounds__(256) void haar_dwt_async(
    const float* __restrict__ x, float* __restrict__ out) {
  __shared__ float tile[2][16 * 512];  // 2 x 32 KB of the 320 KB WGP LDS

  const unsigned t  = threadIdx.x;
  const unsigned g  = blockIdx.x & 7u;   // tile-group within plane (4 tiles each)
  const unsigned p  = blockIdx.x >> 3;   // (b,c) plane, 0..255

  const unsigned c2    = t & 127u;       // output col pair (cols 2*c2, 2*c2+1)
  const unsigned rquad = (t >> 7) << 2;  // output rows rquad..rquad+3 of tile

  // Issue one tile's 2048 B128 chunks (8 per thread), global -> LDS, NT hint.
  auto issue = [&](unsigned it, unsigned buf) {
    const float* gbase =
        x + ((size_t)p << 18) + ((size_t)((g << 2) + it) << 13);
#pragma unroll
    for (int k = 0; k < 8; ++k) {
      unsigned chunk = ((unsigned)k << 8) + t;
      __builtin_amdgcn_global_load_async_to_lds_b128(
          (g_b128_ptr)(gbase + ((size_t)chunk << 2)),
          (l_b128_ptr)(&tile[buf][chunk << 2]),
          /*offset=*/0, /*cpol=*/1 /*TH_LOAD_NT*/);
    }
  };

  // Butterfly one tile out of an LDS buffer into the four subbands.
  auto compute = [&](unsigned it, unsigned buf) {
    const unsigned tb = (g << 2) + it;
    const size_t pos_base = ((size_t)p << 16) + ((size_t)tb << 11);
#pragma unroll
    for (unsigned r4 = 0; r4 < 4; ++r4) {
      const unsigned r = rquad + r4;  // output row within tile (0..7)
      const float* le = &tile[buf][(2u * r) * 512u + (c2 << 2)];
      v4f e = *(const v4f*)le;          // {x1a, x3a, x1b, x3b}
      v4f o = *(const v4f*)(le + 512);  // {x2a, x4a, x2b, x4b}

      float x1a = e.x * 0.5f, x3a = e.y * 0.5f, x1b = e.z * 0.5f, x3b = e.w * 0.5f;
      float x2a = o.x * 0.5f, x4a = o.y * 0.5f, x2b = o.z * 0.5f, x4b = o.w * 0.5f;

      v2f ll = {  x1a + x2a + x3a + x4a,  x1b + x2b + x3b + x4b };
      v2f lh = { -x1a + x2a - x3a + x4a, -x1b + x2b - x3b + x4b };
      v2f hl = { -x1a - x2a + x3a + x4a, -x1b - x2b + x3b + x4b };
      v2f hh = {  x1a - x2a - x3a + x4a,  x1b - x2b - x3b + x4b };

      size_t pos = pos_base + ((size_t)r << 8) + (c2 << 1);
      __builtin_nontemporal_store(ll, (v2f*)(out + pos));
      __builtin_nontemporal_store(lh, (v2f*)(out + (size_t)SUB_ELEMS + pos));
      __builtin_nontemporal_store(hl, (v2f*)(out + 2u * (size_t)SUB_ELEMS + pos));
      __builtin_nontemporal_store(hh, (v2f*)(out + 3u * (size_t)SUB_ELEMS + pos));
    }
  };

  issue(0, 0);
#pragma unroll
  for (unsigned it = 0; it < 3; ++it) {
    issue(it + 1, (it + 1) & 1u);  // prefetch next tile into the other buffer
    // Own 8 newest asyncs (tile it+1) may remain outstanding; the 8 for tile
    // `it` (issued earlier, retired in order) must be done:
    wait_asynccnt<8>();
    __syncthreads();               // all waves' tile-`it` data visible in LDS
    compute(it, it & 1u);
    __syncthreads();               // buffer (it&1) free for reuse next round
  }
  wait_asynccnt<0>();
  __syncthreads();
  compute(3, 1);
}

#else  // !HAVE_ASYNC_LDS

// ---------------------------------------------------------------------------
// Fallback: direct registers-only butterfly. 2 x B128 NT loads (even/odd
// row), 4 x B64 NT stores. Fully coalesced.
// ---------------------------------------------------------------------------
__global__ __launch_bounds__(256) void haar_dwt_direct(
    const float* __restrict__ x, float* __restrict__ out) {
  unsigned tid  = blockIdx.x * 256u + threadIdx.x;
  unsigned j2   = tid & 127u;
  unsigned rest = tid >> 7;
  unsigned i    = rest & 255u;
  unsigned p    = rest >> 8;

  const float* rowe = x + ((size_t)p << 18) + ((size_t)(2u * i) << 9) + (j2 << 2);
  v4f e = __builtin_nontemporal_load((const v4f*)rowe);
  v4f o = __builtin_nontemporal_load((const v4f*)(rowe + 512));

  float x1a = e.x * 0.5f, x3a = e.y * 0.5f, x1b = e.z * 0.5f, x3b = e.w * 0.5f;
  float x2a = o.x * 0.5f, x4a = o.y * 0.5f, x2b = o.z * 0.5f, x4b = o.w * 0.5f;

  v2f ll = {  x1a + x2a + x3a + x4a,  x1b + x2b + x3b + x4b };
  v2f lh = { -x1a + x2a - x3a + x4a, -x1b + x2b - x3b + x4b };
  v2f hl = { -x1a - x2a + x3a + x4a, -x1b - x2b + x3b + x4b };
  v2f hh = {  x1a - x2a - x3a + x4a,  x1b - x2b - x3b + x4b };

  size_t pos = ((size_t)p << 16) + ((size_t)i << 8) + (j2 << 1);
  __builtin_nontemporal_store(ll, (v2f*)(out + pos));
  __builtin_nontemporal_store(lh, (v2f*)(out + (size_t)SUB_ELEMS + pos));
  __builtin_nontemporal_store(hl, (v2f*)(out + 2u * (size_t)SUB_ELEMS + pos));
  __builtin_nontemporal_store(hh, (v2f*)(out + 3u * (size_t)SUB_ELEMS + pos));
}

#endif  // HAVE_ASYNC_LDS

extern "C" void kernel_launch(void* const* d_in, const int* in_sizes, int n_in,
                              void* d_out, int out_size, void* d_ws, size_t ws_size,
                              hipStream_t stream) {
  const float* x = (const float*)d_in[0];
  float* out = (float*)d_out;
  (void)in_sizes; (void)n_in; (void)out_size; (void)d_ws; (void)ws_size;

#ifdef HAVE_ASYNC_LDS
  // 256 planes x 8 tile-groups; 256 threads (8 wave32s), 64 KB LDS each.
  haar_dwt_async<<<256 * 8, 256, 0, stream>>>(x, out);
#else
  haar_dwt_direct<<<(SUB_ELEMS / 2u) / 256u, 256, 0, stream>>>(x, out);
#endif
}